// NeuralODE_76175539962212
// MI455X (gfx1250) — compile-verified
//
#include <hip/hip_runtime.h>
#include <cmath>

#define GEOM  3
#define CODE  64
#define HID   512
#define SDIM  68          // GEOM + 1 + CODE
#define SP    72          // padded LDS stride for s
#define HP    516         // padded LDS stride for h (516 % 64 == 4 -> conflict-free column reads)
#define STEPS 64          // fixed-step RK4
#define MT    32          // samples per workgroup (2 WMMA M-tiles)

typedef float v2f __attribute__((ext_vector_type(2)));
typedef float v8f __attribute__((ext_vector_type(8)));

__device__ __forceinline__ v8f wmma_f32(v2f a, v2f b, v8f c) {
  return __builtin_amdgcn_wmma_f32_16x16x4_f32(false, a, false, b, (short)0, c,
                                               false, false);
}

// Branch-free tanh: native V_TANH_F32 if available, else exp/rcp (both ~1-2 ulp)
__device__ __forceinline__ float fast_tanh(float x) {
#if __has_builtin(__builtin_amdgcn_tanhf)
  return __builtin_amdgcn_tanhf(x);
#else
  float xc = fminf(fmaxf(x, -15.0f), 15.0f);
  float e  = __expf(2.0f * xc);                      // v_exp_f32
  return (e - 1.0f) * __builtin_amdgcn_rcpf(e + 1.0f);
#endif
}

// A fragment (16x4 fp32): lanes 0-15 -> M=lane, K=k0,k0+1 ; lanes 16-31 -> K=k0+2,k0+3
__device__ __forceinline__ v2f a_frag(const float* buf, int stride, int k0, int lane) {
  int m = lane & 15;
  int k = k0 + ((lane >> 4) << 1);
  const float* p = buf + m * stride + k;
  v2f a; a.x = p[0]; a.y = p[1];
  return a;
}

// B fragment (4x16 fp32) from row-major W[K][ldn]
__device__ __forceinline__ v2f b_frag(const float* __restrict__ W, int ldn,
                                      int k0, int n0, int lane) {
  int n = n0 + (lane & 15);
  int k = k0 + ((lane >> 4) << 1);
  v2f b; b.x = W[k * ldn + n]; b.y = W[(k + 1) * ldn + n];
  return b;
}

// B fragment from a Kx3 weight, zero-padded to 16 columns.
// Branch-free: clamp address (always valid) and mask with cndmask-multiply.
__device__ __forceinline__ v2f b_frag3(const float* __restrict__ W, int k0, int lane) {
  int n   = lane & 15;
  int nc  = (n < GEOM) ? n : 0;
  float m = (n < GEOM) ? 1.0f : 0.0f;
  int k = k0 + ((lane >> 4) << 1);
  v2f b;
  b.x = W[k * GEOM + nc] * m;
  b.y = W[(k + 1) * GEOM + nc] * m;
  return b;
}

// One wave: 32 x 64 slice (2 M-tiles x 4 N-tiles) of A(32xK) @ W(KxHID) + bias.
// One B fragment feeds two WMMAs -> 2 global b32 loads per 2 WMMA.
__device__ __forceinline__ void dense_acc2(const float* Ash, int aStride, int K,
                                           const float* __restrict__ W,
                                           const float* __restrict__ bias,
                                           v8f acc[2][4], int lane, int wave) {
  const int n0 = wave * 64;
#pragma unroll
  for (int t = 0; t < 4; ++t) {
    float bv = bias[n0 + t * 16 + (lane & 15)];
#pragma unroll
    for (int i = 0; i < 8; ++i) { acc[0][t][i] = bv; acc[1][t][i] = bv; }
  }
#pragma unroll 2
  for (int k0 = 0; k0 < K; k0 += 4) {
    v2f a0 = a_frag(Ash,                aStride, k0, lane);   // M rows 0-15
    v2f a1 = a_frag(Ash + 16 * aStride, aStride, k0, lane);   // M rows 16-31
#pragma unroll
    for (int t = 0; t < 4; ++t) {
      v2f b = b_frag(W, HID, k0, n0 + t * 16, lane);
      acc[0][t] = wmma_f32(a0, b, acc[0][t]);
      acc[1][t] = wmma_f32(a1, b, acc[1][t]);
    }
  }
}

// C layout: VGPR r -> M = r (lanes 0-15) / r+8 (lanes 16-31), N = n0 + (lane&15)
__device__ __forceinline__ void store_tanh2(float* H, v8f acc[2][4], int lane, int wave) {
  const int n0 = wave * 64;
  const int nn = lane & 15;
  const int mh = (lane >> 4) << 3;
#pragma unroll
  for (int mt = 0; mt < 2; ++mt)
#pragma unroll
    for (int t = 0; t < 4; ++t)
#pragma unroll
      for (int r = 0; r < 8; ++r)
        H[(mt * 16 + mh + r) * HP + n0 + t * 16 + nn] = fast_tanh(acc[mt][t][r]);
}

// 32x512 @ 512x3 (+bias), split-K over 8 waves, deterministic LDS tree reduce
__device__ __forceinline__ void skinny3(const float* H, const float* __restrict__ W,
                                        const float* __restrict__ b3,
                                        float* dst /*32x4*/, float* pbuf,
                                        int lane, int wave, int tid) {
  v8f c0 = {}, c1 = {};
  const int kbeg = wave * 64;
  for (int k0 = kbeg; k0 < kbeg + 64; k0 += 4) {
    v2f b  = b_frag3(W, k0, lane);
    v2f a0 = a_frag(H,           HP, k0, lane);
    v2f a1 = a_frag(H + 16 * HP, HP, k0, lane);
    c0 = wmma_f32(a0, b, c0);
    c1 = wmma_f32(a1, b, c1);
  }
  const int nn = lane & 15;
  const int mh = (lane >> 4) << 3;
#pragma unroll
  for (int r = 0; r < 8; ++r) {
    pbuf[(wave * 2 + 0) * 256 + (mh + r) * 16 + nn] = c0[r];
    pbuf[(wave * 2 + 1) * 256 + (mh + r) * 16 + nn] = c1[r];
  }
  __syncthreads();
  if (tid < MT * GEOM) {                 // 96 threads: one (m, n) each
    int m = tid / GEOM, n = tid % GEOM;
    int mt = m >> 4, mr = m & 15;
    float s = b3[n];
#pragma unroll
    for (int w = 0; w < 8; ++w) s += pbuf[(w * 2 + mt) * 256 + mr * 16 + n];
    dst[m * 4 + n] = s;
  }
  __syncthreads();
}

__global__ __launch_bounds__(256) void node_rk4_kernel(
    const float* __restrict__ T,   const float* __restrict__ code,
    const float* __restrict__ pW1, const float* __restrict__ pb1,
    const float* __restrict__ pW2, const float* __restrict__ pb2,
    const float* __restrict__ oW1, const float* __restrict__ ob1,
    const float* __restrict__ oW2, const float* __restrict__ ob2,
    const float* __restrict__ oW3, const float* __restrict__ ob3,
    float* __restrict__ out) {
  __shared__ float sbuf[MT * SP];        // [yg(3) | t | code(64)] per row
  __shared__ float hbuf[MT * HP];        // hidden activations (h1 reused as h2)
  __shared__ float pbuf[16 * 256];       // split-K partial tiles
  __shared__ float ybuf[MT * 4];         // current y
  __shared__ float kacc[MT * 4];         // RK4 accumulator
  __shared__ float dybuf[MT * 4];        // latest f(y,t)

  const int tid  = threadIdx.x;
  const int lane = tid & 31;
  const int wave = tid >> 5;
  const int r0   = blockIdx.x * MT;

  for (int i = tid; i < MT * CODE; i += 256) {
    int m = i >> 6, j = i & 63;
    sbuf[m * SP + (GEOM + 1) + j] = code[(r0 + m) * CODE + j];
  }
  if (tid < MT * 4) { kacc[tid] = 0.0f; dybuf[tid] = 0.0f; }
  __syncthreads();

  // ---- init: h = tanh(Tpad(32x4) @ pW1 + pb1), one WMMA per (M-tile, N-tile)
  {
    const int n0 = wave * 64;
    v8f acc[2][4];
    const int m = lane & 15;
    v2f a0, a1;
    if (lane < 16) {
      a0.x = T[(r0 + m) * GEOM + 0];      a0.y = T[(r0 + m) * GEOM + 1];
      a1.x = T[(r0 + 16 + m) * GEOM + 0]; a1.y = T[(r0 + 16 + m) * GEOM + 1];
    } else {
      a0.x = T[(r0 + m) * GEOM + 2];      a0.y = 0.0f;
      a1.x = T[(r0 + 16 + m) * GEOM + 2]; a1.y = 0.0f;
    }
    // branch-free guard for pW1's missing K row 3: clamp row, mask value
    const int   k   = (lane >> 4) << 1;
    const int   k1  = k + 1;
    const int   k1c = (k1 < GEOM) ? k1 : 0;
    const float k1m = (k1 < GEOM) ? 1.0f : 0.0f;
#pragma unroll
    for (int t = 0; t < 4; ++t) {
      const int nt = n0 + t * 16;
      float bv = pb1[nt + (lane & 15)];
      v8f c;
#pragma unroll
      for (int i = 0; i < 8; ++i) c[i] = bv;
      const int n = nt + (lane & 15);
      v2f b;
      b.x = pW1[k * HID + n];
      b.y = pW1[k1c * HID + n] * k1m;
      acc[0][t] = wmma_f32(a0, b, c);
      acc[1][t] = wmma_f32(a1, b, c);
    }
    store_tanh2(hbuf, acc, lane, wave);
  }
  __syncthreads();
  skinny3(hbuf, pW2, pb2, ybuf, pbuf, lane, wave, tid);   // y0 = h @ pW2 + pb2

  // ---- fixed-step RK4 on y (code channels have zero derivative)
  const float dt = 1.0f / (float)STEPS;
  for (int step = 0; step < STEPS; ++step) {
    const float t0 = dt * (float)step;
#pragma unroll 1
    for (int s = 0; s < 4; ++s) {
      const float ACO[4] = {0.0f, 0.5f, 0.5f, 1.0f};
      const float BCO[4] = {1.0f, 2.0f, 2.0f, 1.0f};
      const float ac = ACO[s];
      if (tid < MT * 4) {
        int m = tid >> 2, j = tid & 3;
        if (j < GEOM) sbuf[m * SP + j] = ybuf[m * 4 + j] + ac * dt * dybuf[m * 4 + j];
        else          sbuf[m * SP + GEOM] = t0 + ac * dt;
      }
      __syncthreads();
      v8f acc[2][4];
      dense_acc2(sbuf, SP, SDIM, oW1, ob1, acc, lane, wave);  // 17 K-iters
      store_tanh2(hbuf, acc, lane, wave);
      __syncthreads();
      dense_acc2(hbuf, HP, HID, oW2, ob2, acc, lane, wave);   // 128 K-iters
      __syncthreads();                                        // all h1 reads done
      store_tanh2(hbuf, acc, lane, wave);                     // overwrite with h2
      __syncthreads();
      skinny3(hbuf, oW3, ob3, dybuf, pbuf, lane, wave, tid);  // dy -> dybuf
      if (tid < MT * 4) {
        int m = tid >> 2, j = tid & 3;
        if (j < GEOM) kacc[m * 4 + j] += BCO[s] * dybuf[m * 4 + j];
      }
      __syncthreads();
    }
    if (tid < MT * 4) {
      int m = tid >> 2, j = tid & 3;
      if (j < GEOM) ybuf[m * 4 + j] += (dt / 6.0f) * kacc[m * 4 + j];
      kacc[tid] = 0.0f;
    }
    __syncthreads();
  }

  if (tid < MT * GEOM) {
    int m = tid / GEOM, j = tid % GEOM;
    out[(r0 + m) * GEOM + j] = ybuf[m * 4 + j];
  }
}

extern "C" void kernel_launch(void* const* d_in, const int* in_sizes, int n_in,
                              void* d_out, int out_size, void* d_ws, size_t ws_size,
                              hipStream_t stream) {
  const float* T    = (const float*)d_in[0];
  const float* code = (const float*)d_in[1];
  const float* pW1  = (const float*)d_in[2];
  const float* pb1  = (const float*)d_in[3];
  const float* pW2  = (const float*)d_in[4];
  const float* pb2  = (const float*)d_in[5];
  const float* oW1  = (const float*)d_in[6];
  const float* ob1  = (const float*)d_in[7];
  const float* oW2  = (const float*)d_in[8];
  const float* ob2  = (const float*)d_in[9];
  const float* oW3  = (const float*)d_in[10];
  const float* ob3  = (const float*)d_in[11];
  float* out = (float*)d_out;

  const int N = in_sizes[0] / GEOM;
  const int blocks = N / MT;
  hipLaunchKernelGGL(node_rk4_kernel, dim3(blocks), dim3(256), 0, stream,
                     T, code, pW1, pb1, pW2, pb2, oW1, ob1, oW2, ob2, oW3, ob3, out);
}